// PFNet5_41034117546322
// MI455X (gfx1250) — compile-verified
//
#include <hip/hip_runtime.h>
#include <hip/hip_bf16.h>
#include <climits>

// ---------------------------------------------------------------------------
// PFNet graph-U-Net forward for MI455X (gfx1250, wave32, WMMA f32 16x16x4)
// Register-blocked WMMA GEMM: one wave = 16 x (NB*16) output strip, A fragment
// loaded once per K-step and reused across NB back-to-back WMMAs.
// ---------------------------------------------------------------------------

typedef float v2f __attribute__((ext_vector_type(2)));
typedef float v8f __attribute__((ext_vector_type(8)));

#define THREADS 256
#define NNODE 6000
#define NEDGE 96000
#define HID 32
#define KP1 1200
#define KP2 240
#define KP3 48

// ----------------------------- utility kernels -----------------------------

__global__ void k_zero(float* __restrict__ p, size_t count) {
  size_t i = (size_t)blockIdx.x * blockDim.x + threadIdx.x;
  size_t stride = (size_t)gridDim.x * blockDim.x;
  for (; i < count; i += stride) p[i] = 0.0f;
}

// A[dst, src] += 1  (edge_index row0 = src, row1 = dst)
__global__ void k_build_adj(const int* __restrict__ ei, float* __restrict__ A,
                            int e, int n) {
  int t = blockIdx.x * blockDim.x + threadIdx.x;
  if (t >= e) return;
  int src = ei[t];
  int dst = ei[e + t];
  atomicAdd(&A[(size_t)dst * n + src], 1.0f);
}

// dinv[i] = rsqrt(rowsum(A)_i + 2)   (deg of A + 2I; always > 0)
__global__ void k_degree(const float* __restrict__ A, float* __restrict__ dinv,
                         int n) {
  __shared__ float red[THREADS];
  int row = blockIdx.x;
  const float* Ar = A + (size_t)row * n;
  float s = 0.0f;
  for (int j = threadIdx.x; j < n; j += blockDim.x) s += Ar[j];
  red[threadIdx.x] = s;
  __syncthreads();
  for (int st = THREADS / 2; st > 0; st >>= 1) {
    if (threadIdx.x < st) red[threadIdx.x] += red[threadIdx.x + st];
    __syncthreads();
  }
  if (threadIdx.x == 0) dinv[row] = rsqrtf(red[0] + 2.0f);
}

// Z[i,o] = dinv[i] * sum_d X[i,d] * W[d,o]
__global__ void k_xw_scale(const float* __restrict__ X, const float* __restrict__ W,
                           const float* __restrict__ dinv, float* __restrict__ Z,
                           int n, int din, int dout) {
  int t = blockIdx.x * blockDim.x + threadIdx.x;
  if (t >= n * dout) return;
  int i = t / dout, o = t % dout;
  float s = 0.0f;
  for (int d = 0; d < din; ++d) s += X[i * din + d] * W[d * dout + o];
  Z[t] = dinv[i] * s;
}

// out[i,o] = dinv[i] * (Y[i,o] + 2*Z[i,o]) + b[o]   (+ optional ReLU)
__global__ void k_gcn_final(const float* __restrict__ Y, const float* __restrict__ Z,
                            const float* __restrict__ dinv, const float* __restrict__ bvec,
                            float* __restrict__ out, int n, int dout, int relu) {
  int t = blockIdx.x * blockDim.x + threadIdx.x;
  if (t >= n * dout) return;
  int i = t / dout, o = t % dout;
  float v = dinv[i] * (Y[t] + 2.0f * Z[t]) + bvec[o];
  if (relu) v = v > 0.0f ? v : 0.0f;
  out[t] = v;
}

// --------------------------- WMMA fp32 GEMM -------------------------------
// C(MxN) = A(MxK) @ B      NT=false: B is KxN row-major (ldb = row stride)
//                           NT=true : B is NxK row-major (C = A * B^T)
// One wave computes a 16 x (NB*16) strip: NB independent accumulators share
// one A fragment per K-step (A traffic / NB, NB back-to-back v_wmma with no
// D->A/B hazard). M, K multiples of 16; N ragged via clamped B loads +
// predicated stores (EXEC stays all-ones through every WMMA).
template <bool NT, int NB>
__global__ void k_gemm_wmma(const float* __restrict__ A, const float* __restrict__ B,
                            float* __restrict__ C, int M, int N, int K,
                            int lda, int ldb, int ldc, int groupsN) {
  int wave = blockIdx.x * (blockDim.x >> 5) + (threadIdx.x >> 5);
  int lane = threadIdx.x & 31;
  int tilesM = M >> 4;
  if (wave >= tilesM * groupsN) return;  // wave-uniform: EXEC stays all-ones
  int tm = wave / groupsN, tg = wave % groupsN;
  int row0 = tm << 4, col0 = tg * (NB << 4);

  int mr = lane & 15;            // A: lanes 0-15 & 16-31 both cover M=0..15
  int koff = (lane >> 4) << 1;   // lanes 16-31 hold K+2, K+3
  const float* Arow = A + (size_t)(row0 + mr) * lda;

  const float* Bbase[NB];
#pragma unroll
  for (int j = 0; j < NB; ++j) {
    int bcol = col0 + (j << 4) + mr;
    if (bcol > N - 1) bcol = N - 1;  // clamp (garbage cols never stored)
    Bbase[j] = NT ? (B + (size_t)bcol * ldb) : (B + bcol);
  }

  v8f acc[NB];
#pragma unroll
  for (int j = 0; j < NB; ++j) acc[j] = v8f{0.f, 0.f, 0.f, 0.f, 0.f, 0.f, 0.f, 0.f};

  for (int k = 0; k < K; k += 4) {
    __builtin_prefetch(Arow + k + 64, 0, 0);  // global_prefetch_b8 on gfx1250
    v2f a;
    a.x = Arow[k + koff];
    a.y = Arow[k + koff + 1];
#pragma unroll
    for (int j = 0; j < NB; ++j) {
      v2f b;
      if (NT) {
        b.x = Bbase[j][k + koff];
        b.y = Bbase[j][k + koff + 1];
      } else {
        b.x = Bbase[j][(size_t)(k + koff) * ldb];
        b.y = Bbase[j][(size_t)(k + koff + 1) * ldb];
      }
      acc[j] = __builtin_amdgcn_wmma_f32_16x16x4_f32(false, a, false, b,
                                                     (short)0, acc[j], false, false);
    }
  }
  // C/D layout: VGPR i -> row i (lanes 0-15) / row i+8 (lanes 16-31)
  int rbase = row0 + ((lane >> 4) << 3);
#pragma unroll
  for (int j = 0; j < NB; ++j) {
    int c = col0 + (j << 4) + (lane & 15);
    if (c < N) {
      for (int i = 0; i < 8; ++i)
        C[(size_t)(rbase + i) * ldc + c] = acc[j][i];
    }
  }
}

// ------------------------------- pooling ----------------------------------

__global__ void k_pnorm(const float* __restrict__ p, float* __restrict__ out) {
  float s = 0.0f;
  for (int d = 0; d < HID; ++d) s += p[d] * p[d];
  *out = sqrtf(s);
}

__global__ void k_score(const float* __restrict__ h, const float* __restrict__ p,
                        const float* __restrict__ pnorm, float* __restrict__ score,
                        int n) {
  int i = blockIdx.x * blockDim.x + threadIdx.x;
  if (i >= n) return;
  float s = 0.0f;
  for (int d = 0; d < HID; ++d) s += h[i * HID + d] * p[d];
  score[i] = tanhf(s / pnorm[0]);
}

// Single-block bitonic top-k: descending score, ties -> ascending index
// (matches jax.lax.top_k). Pad sentinel -2.0 < tanh range.
#define SORT_CAP 8192
__global__ void k_topk(const float* __restrict__ score, int n, int npad, int k,
                       int* __restrict__ perm, float* __restrict__ vals) {
  __shared__ float ss[SORT_CAP];
  __shared__ int sidx[SORT_CAP];
  for (int i = threadIdx.x; i < npad; i += blockDim.x) {
    if (i < n) { ss[i] = score[i]; sidx[i] = i; }
    else       { ss[i] = -2.0f;    sidx[i] = INT_MAX; }
  }
  __syncthreads();
  for (int size = 2; size <= npad; size <<= 1) {
    for (int stride = size >> 1; stride > 0; stride >>= 1) {
      for (int t = threadIdx.x; t < (npad >> 1); t += blockDim.x) {
        int i = 2 * t - (t & (stride - 1));
        int j = i + stride;
        float vi = ss[i], vj = ss[j];
        int ii = sidx[i], ij = sidx[j];
        bool iFirst = (vi > vj) || (vi == vj && ii < ij);
        bool up = ((i & size) == 0);
        if (up ? !iFirst : iFirst) {
          ss[i] = vj; ss[j] = vi;
          sidx[i] = ij; sidx[j] = ii;
        }
      }
      __syncthreads();
    }
  }
  for (int i = threadIdx.x; i < k; i += blockDim.x) {
    perm[i] = sidx[i];
    vals[i] = ss[i];
  }
}

// hp[i,:] = h[perm[i],:] * vals[i]
__global__ void k_pool_h(const float* __restrict__ h, const int* __restrict__ perm,
                         const float* __restrict__ vals, float* __restrict__ hp,
                         int k) {
  int t = blockIdx.x * blockDim.x + threadIdx.x;
  if (t >= k * HID) return;
  int i = t / HID, d = t % HID;
  hp[t] = h[(size_t)perm[i] * HID + d] * vals[i];
}

// Rg[i,j] = (A + I)[perm[i], j]
__global__ void k_gather_rows(const float* __restrict__ A, const int* __restrict__ perm,
                              float* __restrict__ Rg, int k, int n) {
  int t = blockIdx.x * blockDim.x + threadIdx.x;
  if (t >= k * n) return;
  int i = t / n, j = t % n;
  int pi = perm[i];
  Rg[t] = A[(size_t)pi * n + j] + (j == pi ? 1.0f : 0.0f);
}

// Cg[c,j] = (A + I)[j, perm[c]]   (columns gathered as rows -> NT GEMM)
__global__ void k_gather_cols(const float* __restrict__ A, const int* __restrict__ perm,
                              float* __restrict__ Cg, int k, int n) {
  int t = blockIdx.x * blockDim.x + threadIdx.x;
  if (t >= k * n) return;
  int c = t / n, j = t % n;
  int pc = perm[c];
  Cg[t] = A[(size_t)j * n + pc] + (j == pc ? 1.0f : 0.0f);
}

__global__ void k_zero_diag(float* __restrict__ A, int n) {
  int i = blockIdx.x * blockDim.x + threadIdx.x;
  if (i < n) A[(size_t)i * n + i] = 0.0f;
}

// xbuf[perm[i],:] += h[i,:]   (unpool: zeros.at[perm].set(h) then add)
__global__ void k_unpool_add(float* __restrict__ xbuf, const int* __restrict__ perm,
                             const float* __restrict__ h, int k) {
  int t = blockIdx.x * blockDim.x + threadIdx.x;
  if (t >= k * HID) return;
  int i = t / HID, d = t % HID;
  xbuf[(size_t)perm[i] * HID + d] += h[t];
}

// column-0 batchnorm + sigmoid (biased variance), in place on (n,4) output
__global__ void k_bn_sigmoid(float* __restrict__ out, int n,
                             const float* __restrict__ g, const float* __restrict__ b) {
  __shared__ float rs[1024];
  __shared__ float rq[1024];
  float s = 0.f, q = 0.f;
  for (int i = threadIdx.x; i < n; i += blockDim.x) {
    float v = out[i * 4];
    s += v; q += v * v;
  }
  rs[threadIdx.x] = s; rq[threadIdx.x] = q;
  __syncthreads();
  for (int st = 512; st > 0; st >>= 1) {
    if (threadIdx.x < st) {
      rs[threadIdx.x] += rs[threadIdx.x + st];
      rq[threadIdx.x] += rq[threadIdx.x + st];
    }
    __syncthreads();
  }
  float mean = rs[0] / n;
  float var = rq[0] / n - mean * mean;
  float inv = rsqrtf(var + 1e-5f);
  float gg = g[0], bb = b[0];
  for (int i = threadIdx.x; i < n; i += blockDim.x) {
    float v = out[i * 4];
    float z = gg * (v - mean) * inv + bb;
    out[i * 4] = 1.0f / (1.0f + expf(-z));
  }
}

// ----------------------------- host helpers --------------------------------

static inline int cdiv(int a, int b) { return (a + b - 1) / b; }

// out = relu?( Dinv (A + 2I) Dinv (X W) + b )
// GCN GEMM uses NB=2 (N<=32 -> exactly one column group).
static void run_gcn(const float* Aadj, const float* dinv, const float* X,
                    const float* Wt, const float* bvec, float* outp,
                    float* Z, float* Y, int n, int din, int dout, bool relu,
                    hipStream_t s) {
  k_xw_scale<<<cdiv(n * dout, THREADS), THREADS, 0, s>>>(X, Wt, dinv, Z, n, din, dout);
  int tilesN = cdiv(dout, 16);
  int groupsN = cdiv(tilesN, 2);
  int total = (n >> 4) * groupsN;
  k_gemm_wmma<false, 2><<<cdiv(total, 8), THREADS, 0, s>>>(Aadj, Z, Y, n, dout, n,
                                                           n, dout, dout, groupsN);
  k_gcn_final<<<cdiv(n * dout, THREADS), THREADS, 0, s>>>(Y, Z, dinv, bvec, outp,
                                                          n, dout, relu ? 1 : 0);
}

// Anew = zero_diag( (A+I)[perm,:] @ (A+I)[:,perm] ), dinvNew from Anew
// Square GEMM uses NB=4 (ragged last group handled by clamp/predication).
static void run_twohop(const float* Aadj, const int* perm, float* Rg, float* Cg,
                       float* Anew, float* dinvNew, int k, int n, hipStream_t s) {
  k_gather_rows<<<cdiv(k * n, THREADS), THREADS, 0, s>>>(Aadj, perm, Rg, k, n);
  k_gather_cols<<<cdiv(k * n, THREADS), THREADS, 0, s>>>(Aadj, perm, Cg, k, n);
  int tilesN = k >> 4;
  int groupsN = cdiv(tilesN, 4);
  int total = (k >> 4) * groupsN;
  k_gemm_wmma<true, 4><<<cdiv(total, 8), THREADS, 0, s>>>(Rg, Cg, Anew, k, k, n,
                                                          n, n, k, groupsN);
  k_zero_diag<<<cdiv(k, THREADS), THREADS, 0, s>>>(Anew, k);
  k_degree<<<k, THREADS, 0, s>>>(Anew, dinvNew, k);
}

static void run_pool(const float* h, const float* pvec, float* pnorm, float* score,
                     int* perm, float* vals, float* hp, int n, int npad, int k,
                     hipStream_t s) {
  k_pnorm<<<1, 1, 0, s>>>(pvec, pnorm);
  k_score<<<cdiv(n, THREADS), THREADS, 0, s>>>(h, pvec, pnorm, score, n);
  k_topk<<<1, 1024, 0, s>>>(score, n, npad, k, perm, vals);
  k_pool_h<<<cdiv(k * HID, THREADS), THREADS, 0, s>>>(h, perm, vals, hp, k);
}

// ------------------------------- entry point -------------------------------

extern "C" void kernel_launch(void* const* d_in, const int* in_sizes, int n_in,
                              void* d_out, int out_size, void* d_ws, size_t ws_size,
                              hipStream_t stream) {
  const float* x    = (const float*)d_in[0];
  const int*   ei   = (const int*)d_in[1];
  const float* wd0  = (const float*)d_in[2];
  const float* bd0  = (const float*)d_in[3];
  const float* wd1  = (const float*)d_in[4];
  const float* bd1  = (const float*)d_in[5];
  const float* wd2  = (const float*)d_in[6];
  const float* bd2  = (const float*)d_in[7];
  const float* wd3  = (const float*)d_in[8];
  const float* bd3  = (const float*)d_in[9];
  const float* pl0  = (const float*)d_in[10];
  const float* pl1  = (const float*)d_in[11];
  const float* pl2  = (const float*)d_in[12];
  const float* wu0  = (const float*)d_in[13];
  const float* bu0  = (const float*)d_in[14];
  const float* wu1  = (const float*)d_in[15];
  const float* bu1  = (const float*)d_in[16];
  const float* wu2  = (const float*)d_in[17];
  const float* bu2  = (const float*)d_in[18];
  const float* bng  = (const float*)d_in[19];
  const float* bnb  = (const float*)d_in[20];
  float* out = (float*)d_out;

  // ---- workspace layout (floats); ~210 MB total ----
  float* base = (float*)d_ws;
  size_t off = 0;
  auto alloc = [&](size_t c) { float* p = base + off; off += c; return p; };
  float* A0    = alloc((size_t)NNODE * NNODE);   // 6000^2
  float* Rg    = alloc((size_t)KP1 * NNODE);     // 1200x6000
  float* Cg    = alloc((size_t)KP1 * NNODE);
  float* A1c   = alloc((size_t)KP1 * KP1);
  float* A2c   = alloc((size_t)KP2 * KP2);
  float* A3c   = alloc((size_t)KP3 * KP3);
  float* dinv0 = alloc(NNODE);
  float* dinv1 = alloc(KP1);
  float* dinv2 = alloc(KP2);
  float* dinv3 = alloc(KP3);
  float* h0    = alloc((size_t)NNODE * HID);
  float* h1    = alloc((size_t)KP1 * HID);
  float* h2    = alloc((size_t)KP2 * HID);
  float* h3    = alloc((size_t)KP3 * HID);
  float* hp    = alloc((size_t)KP1 * HID);       // pooled features (pre-GCN)
  float* hu    = alloc((size_t)KP1 * HID);       // up-path GCN output
  float* Z     = alloc((size_t)NNODE * HID);
  float* Y     = alloc((size_t)NNODE * HID);
  float* score = alloc(NNODE);
  float* vals0 = alloc(KP1);
  float* vals1 = alloc(KP2);
  float* vals2 = alloc(KP3);
  float* pnrm  = alloc(1);
  int* perm0 = (int*)alloc(KP1);
  int* perm1 = (int*)alloc(KP2);
  int* perm2 = (int*)alloc(KP3);

  // ---- build adjacency + normalization ----
  k_zero<<<4096, THREADS, 0, stream>>>(A0, (size_t)NNODE * NNODE);
  k_build_adj<<<cdiv(NEDGE, THREADS), THREADS, 0, stream>>>(ei, A0, NEDGE, NNODE);
  k_degree<<<NNODE, THREADS, 0, stream>>>(A0, dinv0, NNODE);

  // ---- down path ----
  run_gcn(A0, dinv0, x, wd0, bd0, h0, Z, Y, NNODE, 3, HID, true, stream);

  run_pool(h0, pl0, pnrm, score, perm0, vals0, hp, NNODE, 8192, KP1, stream);
  run_twohop(A0, perm0, Rg, Cg, A1c, dinv1, KP1, NNODE, stream);
  run_gcn(A1c, dinv1, hp, wd1, bd1, h1, Z, Y, KP1, HID, HID, true, stream);

  run_pool(h1, pl1, pnrm, score, perm1, vals1, hp, KP1, 2048, KP2, stream);
  run_twohop(A1c, perm1, Rg, Cg, A2c, dinv2, KP2, KP1, stream);
  run_gcn(A2c, dinv2, hp, wd2, bd2, h2, Z, Y, KP2, HID, HID, true, stream);

  run_pool(h2, pl2, pnrm, score, perm2, vals2, hp, KP2, 256, KP3, stream);
  run_twohop(A2c, perm2, Rg, Cg, A3c, dinv3, KP3, KP2, stream);
  run_gcn(A3c, dinv3, hp, wd3, bd3, h3, Z, Y, KP3, HID, HID, true, stream);

  // ---- up path ----
  k_unpool_add<<<cdiv(KP3 * HID, THREADS), THREADS, 0, stream>>>(h2, perm2, h3, KP3);
  run_gcn(A2c, dinv2, h2, wu0, bu0, hu, Z, Y, KP2, HID, HID, true, stream);

  k_unpool_add<<<cdiv(KP2 * HID, THREADS), THREADS, 0, stream>>>(h1, perm1, hu, KP2);
  run_gcn(A1c, dinv1, h1, wu1, bu1, hu, Z, Y, KP1, HID, HID, true, stream);

  k_unpool_add<<<cdiv(KP1 * HID, THREADS), THREADS, 0, stream>>>(h0, perm0, hu, KP1);
  run_gcn(A0, dinv0, h0, wu2, bu2, out, Z, Y, NNODE, HID, 4, false, stream);

  // ---- batchnorm + sigmoid on column 0 ----
  k_bn_sigmoid<<<1, 1024, 0, stream>>>(out, NNODE, bng, bnb);
}